// GCN_interaction_all_69131793596497
// MI455X (gfx1250) — compile-verified
//
#include <hip/hip_runtime.h>
#include <hip/hip_bf16.h>

typedef float v2f __attribute__((ext_vector_type(2)));
typedef float v8f __attribute__((ext_vector_type(8)));

#define GEMM_TILE_K 128

// C[M x 64] = A[M x K] @ W[K x 64] (+ bias[64]) (* mulmat[M x 64])
// One wave per 16-row strip, 4 WMMA col-tiles per wave.
// W chunk staged into LDS via CDNA5 async global->LDS DMA (ASYNCcnt).
__global__ __launch_bounds__(256) void gemm64_wmma_kernel(
    const float* __restrict__ A, const float* __restrict__ W,
    const float* __restrict__ bias, const float* __restrict__ mulmat,
    float* __restrict__ outp, int M, int K)
{
    // paired layout: ldsW[(k>>1)*128 + col*2 + (k&1)]
    __shared__ float ldsW[GEMM_TILE_K * 64];

    const int tid  = threadIdx.x;
    const int wave = tid >> 5;
    const int lane = tid & 31;
    const int hl   = lane >> 4;      // half-wave select
    const int l15  = lane & 15;
    const int rowBase = (blockIdx.x * 8 + wave) * 16;

    int rowA = rowBase + l15;
    if (rowA > M - 1) rowA = M - 1;
    const float* aRow = A + (size_t)rowA * (size_t)K;

    v8f acc0 = {}, acc1 = {}, acc2 = {}, acc3 = {};

    for (int k0 = 0; k0 < K; k0 += GEMM_TILE_K) {
        int kc = K - k0; if (kc > GEMM_TILE_K) kc = GEMM_TILE_K;
        // async global->LDS staging of the W chunk (permuted pair layout)
        for (int idx = tid; idx < kc * 64; idx += 256) {
            int r = idx >> 6, c = idx & 63;
            unsigned ldsOff = (unsigned)(uintptr_t)&ldsW[(r >> 1) * 128 + c * 2 + (r & 1)];
            unsigned long long gp =
                (unsigned long long)(W + (size_t)(k0 + r) * 64 + c);
            asm volatile("global_load_async_to_lds_b32 %0, %1, off"
                         :: "v"(ldsOff), "v"(gp)
                         : "memory");
        }
        // wait for this wave's async DMAs, then make LDS visible block-wide
        asm volatile("s_wait_asynccnt 0x0" ::: "memory");
        __syncthreads();
        if (k0 + GEMM_TILE_K < K)
            __builtin_prefetch(aRow + k0 + GEMM_TILE_K, 0, 0);

        for (int kk = 0; kk < kc; kk += 4) {
            // A fragment: lane (hl,l15) holds A[row][kk+2*hl + {0,1}] (streamed once -> NT)
            v2f a = __builtin_nontemporal_load((const v2f*)(aRow + k0 + kk + 2 * hl));
            const int pr = ((kk >> 1) + hl) * 128 + l15 * 2;
            v2f b0, b1, b2, b3;
            { float2 t = *(const float2*)&ldsW[pr +  0]; b0[0] = t.x; b0[1] = t.y; }
            { float2 t = *(const float2*)&ldsW[pr + 32]; b1[0] = t.x; b1[1] = t.y; }
            { float2 t = *(const float2*)&ldsW[pr + 64]; b2[0] = t.x; b2[1] = t.y; }
            { float2 t = *(const float2*)&ldsW[pr + 96]; b3[0] = t.x; b3[1] = t.y; }
            acc0 = __builtin_amdgcn_wmma_f32_16x16x4_f32(false, a, false, b0, (short)0, acc0, false, false);
            acc1 = __builtin_amdgcn_wmma_f32_16x16x4_f32(false, a, false, b1, (short)0, acc1, false, false);
            acc2 = __builtin_amdgcn_wmma_f32_16x16x4_f32(false, a, false, b2, (short)0, acc2, false, false);
            acc3 = __builtin_amdgcn_wmma_f32_16x16x4_f32(false, a, false, b3, (short)0, acc3, false, false);
        }
        __syncthreads();
    }

    // epilogue: C/D layout -> row = rowBase + v + 8*hl, col = 16*t + l15
    for (int v = 0; v < 8; ++v) {
        int rowO = rowBase + v + 8 * hl;
        if (rowO >= M) continue;
        float x0 = acc0[v], x1 = acc1[v], x2 = acc2[v], x3 = acc3[v];
        if (bias) {
            x0 += bias[ 0 + l15]; x1 += bias[16 + l15];
            x2 += bias[32 + l15]; x3 += bias[48 + l15];
        }
        size_t o = (size_t)rowO * 64;
        if (mulmat) {
            x0 *= mulmat[o +  0 + l15]; x1 *= mulmat[o + 16 + l15];
            x2 *= mulmat[o + 32 + l15]; x3 *= mulmat[o + 48 + l15];
        }
        outp[o +  0 + l15] = x0; outp[o + 16 + l15] = x1;
        outp[o + 32 + l15] = x2; outp[o + 48 + l15] = x3;
    }
}

// out[dst[e]*64+d] += t[src[e]*64+d]
__global__ void scatter_plain_kernel(const float* __restrict__ t,
                                     const int* __restrict__ src, const int* __restrict__ dst,
                                     float* __restrict__ outp, int E)
{
    int idx = blockIdx.x * 256 + threadIdx.x;
    if (idx >= E * 64) return;
    int e = idx >> 6, d = idx & 63;
    atomicAdd(&outp[dst[e] * 64 + d], t[src[e] * 64 + d]);
}

// out[dst[e]*stride+d] += t[src[e]*64+d] * sigmoid(remb[rating[e]*64+d])
__global__ void scatter_rating_kernel(const float* __restrict__ t,
                                      const int* __restrict__ src, const int* __restrict__ dst,
                                      const int* __restrict__ rating, const float* __restrict__ remb,
                                      float* __restrict__ outp, int stride, int E)
{
    int idx = blockIdx.x * 256 + threadIdx.x;
    if (idx >= E * 64) return;
    int e = idx >> 6, d = idx & 63;
    float x = remb[rating[e] * 64 + d];
    float s = 1.0f / (1.0f + __expf(-x));
    atomicAdd(&outp[dst[e] * stride + d], t[src[e] * 64 + d] * s);
}

// fused fe5 + re scatter:
//  o4[dst*stride+d] += (w[src*64+d] + re[e*64+d]) * cmat[src*64+d]
//  o5[dst*stride+d] +=  re[e*64+d]               * cmat[src*64+d]
__global__ void scatter_fe5_kernel(const float* __restrict__ w, const float* __restrict__ re,
                                   const float* __restrict__ cmat,
                                   const int* __restrict__ src, const int* __restrict__ dst,
                                   float* __restrict__ o4, float* __restrict__ o5,
                                   int stride, int E)
{
    int idx = blockIdx.x * 256 + threadIdx.x;
    if (idx >= E * 64) return;
    int e = idx >> 6, d = idx & 63;
    int s = src[e], t = dst[e];
    float cu = cmat[s * 64 + d];
    float r  = re[idx];
    atomicAdd(&o4[t * stride + d], (w[s * 64 + d] + r) * cu);
    atomicAdd(&o5[t * stride + d], r * cu);
}

// tmp = (mask*h*cmat + (1-mask)*fe) * cui
__global__ void combine_kernel(const float* __restrict__ h, const float* __restrict__ cmat,
                               const float* __restrict__ fe, const float* __restrict__ mask,
                               const float* __restrict__ cui, float* __restrict__ outp, int R)
{
    int idx = blockIdx.x * 256 + threadIdx.x;
    if (idx >= R * 64) return;
    int r = idx >> 6;
    float m = mask[r];
    outp[idx] = (m * h[idx] * cmat[idx] + (1.0f - m) * fe[idx]) * cui[idx];
}

__global__ void mul_kernel(const float* __restrict__ a, const float* __restrict__ b,
                           float* __restrict__ o, int n)
{
    int idx = blockIdx.x * 256 + threadIdx.x;
    if (idx < n) o[idx] = a[idx] * b[idx];
}

// out[r*stride+d] *= c[r*64+d]
__global__ void finalize_kernel(float* __restrict__ outp, int stride,
                                const float* __restrict__ c, int R)
{
    int idx = blockIdx.x * 256 + threadIdx.x;
    if (idx >= R * 64) return;
    int r = idx >> 6, d = idx & 63;
    outp[r * stride + d] *= c[idx];
}

static inline int cdiv(int a, int b) { return (a + b - 1) / b; }

extern "C" void kernel_launch(void* const* d_in, const int* in_sizes, int n_in,
                              void* d_out, int out_size, void* d_ws, size_t ws_size,
                              hipStream_t stream)
{
    (void)in_sizes; (void)n_in; (void)ws_size;
    const int U = 20000, I = 15000, AD = 2048, D = 64, NC = 2048;
    const int N = U + I;
    const int E_AU = 300000, E_AI = 300000, E_UI = 300000, E_R = 60000;

    const float* feature1   = (const float*)d_in[0];
    const float* feature2   = (const float*)d_in[1];
    const float* feature3   = (const float*)d_in[2];
    // weights: d_in[3..7]
    const float* W_review1  = (const float*)d_in[8];
    const float* W_review2  = (const float*)d_in[9];
    const float* remb1      = (const float*)d_in[10];
    const float* remb2      = (const float*)d_in[11];
    const float* remb3      = (const float*)d_in[12];
    const float* remb4      = (const float*)d_in[13];
    const float* remb5      = (const float*)d_in[14];
    const float* remb6      = (const float*)d_in[15];
    const float* W_aspect   = (const float*)d_in[16];
    const float* W_aspect2  = (const float*)d_in[17];
    const float* W_cnt_u    = (const float*)d_in[18];
    const float* b_cnt_u    = (const float*)d_in[19];
    const float* W_cnt_i    = (const float*)d_in[20];
    const float* b_cnt_i    = (const float*)d_in[21];
    const float* aspect_fe  = (const float*)d_in[22];
    const float* aspect_cau = (const float*)d_in[23];
    const float* aspect_cai = (const float*)d_in[24];
    const float* user_cau   = (const float*)d_in[25];
    const float* user_cui   = (const float*)d_in[26];
    const float* item_cai   = (const float*)d_in[27];
    const float* item_ciu   = (const float*)d_in[28];
    // d_in[29+2k] = user_cui{k+1}, d_in[30+2k] = item_ciu{k+1}
    const float* user_mask  = (const float*)d_in[39];
    const float* item_mask  = (const float*)d_in[40];
    const float* u_cnt      = (const float*)d_in[41];
    const float* i_cnt      = (const float*)d_in[42];
    // d_in[43],[44] = ui_review/iu_review (unused by reference)
    // d_in[45+2k] = ui{k+1}_review, d_in[46+2k] = iu{k+1}_review
    const int* au_src = (const int*)d_in[55];
    const int* au_dst = (const int*)d_in[56];
    const int* ai_src = (const int*)d_in[57];
    const int* ai_dst = (const int*)d_in[58];
    const int* ui_src = (const int*)d_in[59];
    const int* ui_dst = (const int*)d_in[60];
    const int* ui_rat = (const int*)d_in[61];
    const int* iu_src = (const int*)d_in[62];
    const int* iu_dst = (const int*)d_in[63];
    const int* iu_rat = (const int*)d_in[64];
    // d_in[65+4k]=ui{k+1}_src, [66+4k]=ui{k+1}_dst, [67+4k]=iu{k+1}_src, [68+4k]=iu{k+1}_dst

    float* out  = (float*)d_out;
    float* out1 = out;                                  // senti  (N,64)
    float* out2 = out1 + (size_t)N * D;                 // count  (N,64)
    float* out3 = out2 + (size_t)N * D;                 // fe     (N,64)
    float* out4 = out3 + (size_t)N * D;                 // fe5    (N,320)
    float* out5 = out4 + (size_t)N * 5 * D;             // re     (N,320)
    const int ST5 = 5 * D;                              // 320

    float* ws   = (float*)d_ws;
    float* t_au = ws;                                   // AD*D
    float* t_ai = t_au + (size_t)AD * D;                // AD*D
    float* hU   = t_ai + (size_t)AD * D;                // U*D (accumulator / GEMM out)
    float* hI   = hU + (size_t)U * D;                   // I*D
    float* tU   = hI + (size_t)I * D;                   // U*D
    float* tI   = tU + (size_t)U * D;                   // I*D
    float* reU  = tI + (size_t)I * D;                   // E_R*D
    float* reI  = reU + (size_t)E_R * D;                // E_R*D

    (void)hipMemsetAsync(d_out, 0, (size_t)out_size * sizeof(float), stream);
    (void)hipMemsetAsync(hU, 0, (size_t)(U + I) * D * sizeof(float), stream);

    auto gemm = [&](const float* A, const float* W, const float* bias,
                    const float* mm, float* o, int M, int K) {
        int blocks = cdiv(cdiv(M, 16), 8);
        hipLaunchKernelGGL(gemm64_wmma_kernel, dim3(blocks), dim3(256), 0, stream,
                           A, W, bias, mm, o, M, K);
    };

    // ---- stage A: aspect propagation ----
    gemm(aspect_fe, W_aspect,  nullptr, aspect_cau, t_au, AD, D);
    gemm(aspect_fe, W_aspect2, nullptr, aspect_cai, t_ai, AD, D);
    hipLaunchKernelGGL(scatter_plain_kernel, dim3(cdiv(E_AU * 64, 256)), dim3(256), 0, stream,
                       t_au, au_src, au_dst, hU, E_AU);
    hipLaunchKernelGGL(scatter_plain_kernel, dim3(cdiv(E_AI * 64, 256)), dim3(256), 0, stream,
                       t_ai, ai_src, ai_dst, hI, E_AI);

    // ---- stage B/C: sentiment features -> out1 ----
    hipLaunchKernelGGL(combine_kernel, dim3(cdiv(U * 64, 256)), dim3(256), 0, stream,
                       hU, user_cau, feature2, user_mask, user_cui, tU, U);
    hipLaunchKernelGGL(combine_kernel, dim3(cdiv(I * 64, 256)), dim3(256), 0, stream,
                       hI, item_cai, feature2 + (size_t)U * D, item_mask, item_ciu, tI, I);
    hipLaunchKernelGGL(scatter_rating_kernel, dim3(cdiv(E_UI * 64, 256)), dim3(256), 0, stream,
                       tU, ui_src, ui_dst, ui_rat, remb1, out1 + (size_t)U * D, D, E_UI);
    hipLaunchKernelGGL(scatter_rating_kernel, dim3(cdiv(E_UI * 64, 256)), dim3(256), 0, stream,
                       tI, iu_src, iu_dst, iu_rat, remb2, out1, D, E_UI);
    hipLaunchKernelGGL(finalize_kernel, dim3(cdiv(U * 64, 256)), dim3(256), 0, stream,
                       out1, D, user_cui, U);
    hipLaunchKernelGGL(finalize_kernel, dim3(cdiv(I * 64, 256)), dim3(256), 0, stream,
                       out1 + (size_t)U * D, D, item_ciu, I);

    // ---- stage D: count features -> out2 (big WMMA GEMMs, K=2048) ----
    gemm(u_cnt, W_cnt_u, b_cnt_u, nullptr, hU, U, NC);
    gemm(i_cnt, W_cnt_i, b_cnt_i, nullptr, hI, I, NC);
    hipLaunchKernelGGL(combine_kernel, dim3(cdiv(U * 64, 256)), dim3(256), 0, stream,
                       hU, user_cau, feature3, user_mask, user_cui, tU, U);
    hipLaunchKernelGGL(combine_kernel, dim3(cdiv(I * 64, 256)), dim3(256), 0, stream,
                       hI, item_cai, feature3 + (size_t)U * D, item_mask, item_ciu, tI, I);
    hipLaunchKernelGGL(scatter_rating_kernel, dim3(cdiv(E_UI * 64, 256)), dim3(256), 0, stream,
                       tU, ui_src, ui_dst, ui_rat, remb3, out2 + (size_t)U * D, D, E_UI);
    hipLaunchKernelGGL(scatter_rating_kernel, dim3(cdiv(E_UI * 64, 256)), dim3(256), 0, stream,
                       tI, iu_src, iu_dst, iu_rat, remb4, out2, D, E_UI);
    hipLaunchKernelGGL(finalize_kernel, dim3(cdiv(U * 64, 256)), dim3(256), 0, stream,
                       out2, D, user_cui, U);
    hipLaunchKernelGGL(finalize_kernel, dim3(cdiv(I * 64, 256)), dim3(256), 0, stream,
                       out2 + (size_t)U * D, D, item_ciu, I);

    // ---- stage F: plain features -> out3 ----
    hipLaunchKernelGGL(mul_kernel, dim3(cdiv(U * 64, 256)), dim3(256), 0, stream,
                       feature1, user_cui, tU, U * 64);
    hipLaunchKernelGGL(mul_kernel, dim3(cdiv(I * 64, 256)), dim3(256), 0, stream,
                       feature1 + (size_t)U * D, item_ciu, tI, I * 64);
    hipLaunchKernelGGL(scatter_rating_kernel, dim3(cdiv(E_UI * 64, 256)), dim3(256), 0, stream,
                       tU, ui_src, ui_dst, ui_rat, remb5, out3 + (size_t)U * D, D, E_UI);
    hipLaunchKernelGGL(scatter_rating_kernel, dim3(cdiv(E_UI * 64, 256)), dim3(256), 0, stream,
                       tI, iu_src, iu_dst, iu_rat, remb6, out3, D, E_UI);
    hipLaunchKernelGGL(finalize_kernel, dim3(cdiv(U * 64, 256)), dim3(256), 0, stream,
                       out3, D, user_cui, U);
    hipLaunchKernelGGL(finalize_kernel, dim3(cdiv(I * 64, 256)), dim3(256), 0, stream,
                       out3 + (size_t)U * D, D, item_ciu, I);

    // ---- stage E: 5 review channels -> out4 (fe5) and out5 (re) ----
    for (int k = 0; k < 5; ++k) {
        const float* w_k    = (const float*)d_in[3 + k];
        const float* cui_k  = (const float*)d_in[29 + 2 * k];
        const float* ciu_k  = (const float*)d_in[30 + 2 * k];
        const float* ui_rev = (const float*)d_in[45 + 2 * k];
        const float* iu_rev = (const float*)d_in[46 + 2 * k];
        const int* us = (const int*)d_in[65 + 4 * k];
        const int* ud = (const int*)d_in[66 + 4 * k];
        const int* is = (const int*)d_in[67 + 4 * k];
        const int* id = (const int*)d_in[68 + 4 * k];

        gemm(ui_rev, W_review1, nullptr, nullptr, reU, E_R, D);
        gemm(iu_rev, W_review2, nullptr, nullptr, reI, E_R, D);

        // user->item edges: wu = w_k[:U], cmat = user cui_k; dest = item rows
        hipLaunchKernelGGL(scatter_fe5_kernel, dim3(cdiv(E_R * 64, 256)), dim3(256), 0, stream,
                           w_k, reU, cui_k, us, ud,
                           out4 + (size_t)U * ST5 + k * D,
                           out5 + (size_t)U * ST5 + k * D, ST5, E_R);
        // item->user edges: wi = w_k[U:], cmat = item ciu_k; dest = user rows
        hipLaunchKernelGGL(scatter_fe5_kernel, dim3(cdiv(E_R * 64, 256)), dim3(256), 0, stream,
                           w_k + (size_t)U * D, reI, ciu_k, is, id,
                           out4 + k * D, out5 + k * D, ST5, E_R);

        hipLaunchKernelGGL(finalize_kernel, dim3(cdiv(U * 64, 256)), dim3(256), 0, stream,
                           out4 + k * D, ST5, cui_k, U);
        hipLaunchKernelGGL(finalize_kernel, dim3(cdiv(I * 64, 256)), dim3(256), 0, stream,
                           out4 + (size_t)U * ST5 + k * D, ST5, ciu_k, I);
        hipLaunchKernelGGL(finalize_kernel, dim3(cdiv(U * 64, 256)), dim3(256), 0, stream,
                           out5 + k * D, ST5, cui_k, U);
        hipLaunchKernelGGL(finalize_kernel, dim3(cdiv(I * 64, 256)), dim3(256), 0, stream,
                           out5 + (size_t)U * ST5 + k * D, ST5, ciu_k, I);
    }
}